// up_block_2516850835582
// MI455X (gfx1250) — compile-verified
//
#include <hip/hip_runtime.h>

// ---------------------------------------------------------------------------
// up_block for MI455X (gfx1250): fp32 WMMA (V_WMMA_F32_16X16X4_F32) GEMMs,
// LDS-staged one-ring gathers, deterministic two-stage BN reduction.
// ---------------------------------------------------------------------------

typedef float v2f __attribute__((ext_vector_type(2)));
typedef float v8f __attribute__((ext_vector_type(8)));

#define N_COARSE 40962
#define N_FINE   163842
#define IN_CH    128
#define OUT_CH   64
#define SLOPE    0.2f
#define BN_EPS   1e-5f

// LDS row strides chosen so (stride mod 64) == 4 dwords -> the 16 A-rows map
// their 4-dword fragment windows onto all 64 LDS banks exactly once.
#define UP_STRIDE   132   // K=128  (gemm_up)
#define C1_STRIDE   900   // K=896  (conv1)
#define C2_STRIDE   452   // K=448  (conv2)

// ---------------------------------------------------------------------------
// Kernel 1: h[N_COARSE,448] = x1 @ W_up^T + b_up   (flat layout == reshape view)
// grid (2561, 7), block 128 (4 waves; wave w covers cols j*64 + w*16 .. +15)
// ---------------------------------------------------------------------------
__global__ __launch_bounds__(128)
void gemm_up_kernel(const float* __restrict__ x1, const float* __restrict__ Wup,
                    const float* __restrict__ bup, float* __restrict__ h)
{
    __shared__ float As[16 * UP_STRIDE];
    const int rowbase = blockIdx.x << 4;
    const int j       = blockIdx.y;           // 0..6
    const int tid     = threadIdx.x;

    // stage 16 x 128 tile of x1 (clamped rows for the ragged tail)
    for (int g = tid; g < (16 * IN_CH / 4); g += 128) {
        int r = g >> 5;                        // 32 float4 per row
        int q = (g & 31) << 2;
        int row = rowbase + r;
        if (row >= N_COARSE) row = N_COARSE - 1;
        *(float4*)(As + r * UP_STRIDE + q) =
            *(const float4*)(x1 + (size_t)row * IN_CH + q);
    }
    __syncthreads();

    const int lane = tid & 31;
    const int wave = tid >> 5;
    const int ln   = lane & 15;
    const int hi   = lane >> 4;
    const int col  = j * 64 + (wave << 4) + ln;   // output column 0..447

    v8f acc = {};
    const float* Ap = As  + ln * UP_STRIDE       + (hi << 1);
    const float* Bp = Wup + (size_t)col * IN_CH  + (hi << 1);
    for (int k0 = 0; k0 < IN_CH; k0 += 16) {
#pragma unroll
        for (int u = 0; u < 4; ++u) {
            v2f a = *(const v2f*)(Ap + k0 + 4 * u);
            v2f b = *(const v2f*)(Bp + k0 + 4 * u);
            acc = __builtin_amdgcn_wmma_f32_16x16x4_f32(
                false, a, false, b, (short)0, acc, false, false);
        }
    }
    const float bias = bup[col];
#pragma unroll
    for (int r = 0; r < 8; ++r) {
        int row = rowbase + r + (hi << 3);
        if (row < N_COARSE)
            h[(size_t)row * 448 + col] = acc[r] + bias;
    }
}

// ---------------------------------------------------------------------------
// Kernel 2: x1u[N_FINE,64] from top gather + down pair-mean gather.
// ---------------------------------------------------------------------------
__global__ __launch_bounds__(256)
void build_x1u_kernel(const float* __restrict__ h, const int* __restrict__ top,
                      const int* __restrict__ down, float* __restrict__ x1u)
{
    int t = blockIdx.x * 256 + threadIdx.x;
    if (t >= N_FINE * OUT_CH) return;
    int i = t >> 6, c = t & 63;
    float v;
    if (i < N_COARSE) {
        v = h[((size_t)top[i] << 6) + c];
    } else {
        int r = i - N_COARSE;
        int g = 2 * r + (c >= 32 ? 1 : 0);
        int base = (c >= 32) ? (2 * c - 64) : (2 * c);
        size_t o = ((size_t)down[g] << 6) + base;
        v = 0.5f * (h[o] + h[o + 1]);
    }
    x1u[t] = v;
}

// ---------------------------------------------------------------------------
// Kernel 3: conv1  z1[i,c] = sum_j sum_k [x1u|x2][neigh(i,j)][k] * W1[c, j*128+k] + b1[c]
// grid 10241, block 128, dynamic LDS 16*900*4 bytes.
// ---------------------------------------------------------------------------
__global__ __launch_bounds__(128)
void conv1_gemm_kernel(const float* __restrict__ x1u, const float* __restrict__ x2,
                       const int* __restrict__ neigh, const float* __restrict__ W1,
                       const float* __restrict__ b1, float* __restrict__ z)
{
    extern __shared__ float As[];             // [16][C1_STRIDE]
    const int rowbase = blockIdx.x << 4;
    const int tid     = threadIdx.x;

    // stage 16 rows x 896 (7 neighbors x [x1u(64)|x2(64)])
    for (int g = tid; g < (16 * 896 / 4); g += 128) {
        int f  = g << 2;
        int r  = f / 896;
        int o  = f - r * 896;
        int j  = o >> 7;
        int t0 = o & 127;
        int row = rowbase + r;
        if (row >= N_FINE) row = N_FINE - 1;
        int idx = neigh[row * 7 + j];
        const float* src = (t0 < 64) ? (x1u + ((size_t)idx << 6) + t0)
                                     : (x2  + ((size_t)idx << 6) + (t0 - 64));
        *(float4*)(As + r * C1_STRIDE + o) = *(const float4*)src;
    }
    __syncthreads();

    const int lane = tid & 31;
    const int wave = tid >> 5;
    const int ln   = lane & 15;
    const int hi   = lane >> 4;
    const int col  = (wave << 4) + ln;        // 0..63

    v8f acc = {};
    const float* Ap = As + ln * C1_STRIDE  + (hi << 1);
    const float* Bp = W1 + (size_t)col * 896 + (hi << 1);
    for (int k0 = 0; k0 < 896; k0 += 16) {
#pragma unroll
        for (int u = 0; u < 4; ++u) {
            v2f a = *(const v2f*)(Ap + k0 + 4 * u);
            v2f b = *(const v2f*)(Bp + k0 + 4 * u);
            acc = __builtin_amdgcn_wmma_f32_16x16x4_f32(
                false, a, false, b, (short)0, acc, false, false);
        }
    }
    const float bias = b1[col];
#pragma unroll
    for (int r = 0; r < 8; ++r) {
        int row = rowbase + r + (hi << 3);
        if (row < N_FINE)
            z[((size_t)row << 6) + col] = acc[r] + bias;
    }
}

// ---------------------------------------------------------------------------
// Kernel 4: conv2  (same structure, input y[N_FINE,64], K=448)
// ---------------------------------------------------------------------------
__global__ __launch_bounds__(128)
void conv2_gemm_kernel(const float* __restrict__ y, const int* __restrict__ neigh,
                       const float* __restrict__ W2, const float* __restrict__ b2,
                       float* __restrict__ z)
{
    extern __shared__ float As[];             // [16][C2_STRIDE]
    const int rowbase = blockIdx.x << 4;
    const int tid     = threadIdx.x;

    for (int g = tid; g < (16 * 448 / 4); g += 128) {
        int f  = g << 2;
        int r  = f / 448;
        int o  = f - r * 448;
        int j  = o >> 6;
        int t0 = o & 63;
        int row = rowbase + r;
        if (row >= N_FINE) row = N_FINE - 1;
        int idx = neigh[row * 7 + j];
        *(float4*)(As + r * C2_STRIDE + o) =
            *(const float4*)(y + ((size_t)idx << 6) + t0);
    }
    __syncthreads();

    const int lane = tid & 31;
    const int wave = tid >> 5;
    const int ln   = lane & 15;
    const int hi   = lane >> 4;
    const int col  = (wave << 4) + ln;

    v8f acc = {};
    const float* Ap = As + ln * C2_STRIDE  + (hi << 1);
    const float* Bp = W2 + (size_t)col * 448 + (hi << 1);
    for (int k0 = 0; k0 < 448; k0 += 16) {
#pragma unroll
        for (int u = 0; u < 4; ++u) {
            v2f a = *(const v2f*)(Ap + k0 + 4 * u);
            v2f b = *(const v2f*)(Bp + k0 + 4 * u);
            acc = __builtin_amdgcn_wmma_f32_16x16x4_f32(
                false, a, false, b, (short)0, acc, false, false);
        }
    }
    const float bias = b2[col];
#pragma unroll
    for (int r = 0; r < 8; ++r) {
        int row = rowbase + r + (hi << 3);
        if (row < N_FINE)
            z[((size_t)row << 6) + col] = acc[r] + bias;
    }
}

// ---------------------------------------------------------------------------
// BN stats, deterministic: 1024 blocks write per-block per-channel partials
// (fixed-order in-block tree), then one block reduces in fixed order.
// ---------------------------------------------------------------------------
__global__ __launch_bounds__(256)
void stats_partial_kernel(const float* __restrict__ z, float* __restrict__ part)
{
    __shared__ float sh[256 * 8];
    const int tid = threadIdx.x;
    const int gt  = blockIdx.x * 256 + tid;
    const int stride  = 1024 * 256;                 // multiple of 16
    const int totalF4 = (N_FINE * OUT_CH) / 4;
    float s0 = 0, s1 = 0, s2 = 0, s3 = 0, q0 = 0, q1 = 0, q2 = 0, q3 = 0;
    for (int g = gt; g < totalF4; g += stride) {    // each thread owns 4 fixed channels
        float4 v = ((const float4*)z)[g];
        s0 += v.x; q0 += v.x * v.x;
        s1 += v.y; q1 += v.y * v.y;
        s2 += v.z; q2 += v.z * v.z;
        s3 += v.w; q3 += v.w * v.w;
    }
    float* m = sh + tid * 8;
    m[0] = s0; m[1] = s1; m[2] = s2; m[3] = s3;
    m[4] = q0; m[5] = q1; m[6] = q2; m[7] = q3;
    __syncthreads();
    if (tid < 64) {
        int c = tid, tb = c >> 2, comp = c & 3;
        float ss = 0.0f, qq = 0.0f;
        for (int q = 0; q < 16; ++q) {              // fixed order -> deterministic
            int t = tb + (q << 4);                  // threads with t%16 == c/4
            ss += sh[t * 8 + comp];
            qq += sh[t * 8 + 4 + comp];
        }
        part[blockIdx.x * 128 + c]      = ss;
        part[blockIdx.x * 128 + 64 + c] = qq;
    }
}

__global__ __launch_bounds__(256)
void reduce_finalize_kernel(const float* __restrict__ part,
                            const float* __restrict__ gamma,
                            const float* __restrict__ beta,
                            float* __restrict__ scale, float* __restrict__ shift)
{
    __shared__ float ss[4][64], sq[4][64];
    const int tid = threadIdx.x;
    const int c = tid & 63, seg = tid >> 6;
    float s = 0.0f, q = 0.0f;
    for (int b = seg * 256; b < (seg + 1) * 256; ++b) {
        s += part[b * 128 + c];
        q += part[b * 128 + 64 + c];
    }
    ss[seg][c] = s; sq[seg][c] = q;
    __syncthreads();
    if (tid < 64) {
        float S = ((ss[0][tid] + ss[1][tid]) + ss[2][tid]) + ss[3][tid];
        float Q = ((sq[0][tid] + sq[1][tid]) + sq[2][tid]) + sq[3][tid];
        float m  = S * (1.0f / (float)N_FINE);
        float v  = Q * (1.0f / (float)N_FINE) - m * m;
        float rs = rsqrtf(v + BN_EPS);
        float sc = rs * gamma[tid];
        scale[tid] = sc;
        shift[tid] = beta[tid] - m * sc;
    }
}

__global__ __launch_bounds__(256)
void apply_bn_lrelu_kernel(const float* __restrict__ z, const float* __restrict__ scale,
                           const float* __restrict__ shift, float* __restrict__ out)
{
    int g = blockIdx.x * 256 + threadIdx.x;
    const int totalF4 = (N_FINE * OUT_CH) / 4;
    if (g >= totalF4) return;
    float4 v = ((const float4*)z)[g];
    int c0 = (g & 15) << 2;
    float a0 = v.x * scale[c0 + 0] + shift[c0 + 0]; a0 = a0 > 0.0f ? a0 : SLOPE * a0;
    float a1 = v.y * scale[c0 + 1] + shift[c0 + 1]; a1 = a1 > 0.0f ? a1 : SLOPE * a1;
    float a2 = v.z * scale[c0 + 2] + shift[c0 + 2]; a2 = a2 > 0.0f ? a2 : SLOPE * a2;
    float a3 = v.w * scale[c0 + 3] + shift[c0 + 3]; a3 = a3 > 0.0f ? a3 : SLOPE * a3;
    ((float4*)out)[g] = make_float4(a0, a1, a2, a3);
}

// ---------------------------------------------------------------------------
// Workspace layout (bytes); total required = 157,816,320 (~150.5 MB).
//   h   [0, 73403904)                (reused by y after conv1)
//   x1u [73403904, 115347456)        (reused by z2 after conv1)
//   z1  [115347456, 157291008)
//   part[157291008, 157815296)       1024 blocks x 128 floats
//   red [157815296, 157816320)       scale(64) + shift(64)
// ---------------------------------------------------------------------------
extern "C" void kernel_launch(void* const* d_in, const int* in_sizes, int n_in,
                              void* d_out, int out_size, void* d_ws, size_t ws_size,
                              hipStream_t stream)
{
    const float* x1   = (const float*)d_in[0];
    const float* x2   = (const float*)d_in[1];
    const int*   top  = (const int*)d_in[2];
    const int*   down = (const int*)d_in[3];
    const int*   ngh  = (const int*)d_in[4];
    const float* Wup  = (const float*)d_in[5];
    const float* bup  = (const float*)d_in[6];
    const float* W1   = (const float*)d_in[7];
    const float* b1   = (const float*)d_in[8];
    const float* g1   = (const float*)d_in[9];
    const float* be1  = (const float*)d_in[10];
    const float* W2   = (const float*)d_in[11];
    const float* b2   = (const float*)d_in[12];
    const float* g2   = (const float*)d_in[13];
    const float* be2  = (const float*)d_in[14];

    if (ws_size < 157816320u) return;
    char* ws = (char*)d_ws;
    float* h     = (float*)(ws + 0);
    float* x1u   = (float*)(ws + 73403904u);
    float* z1    = (float*)(ws + 115347456u);
    float* part  = (float*)(ws + 157291008u);
    float* scale = (float*)(ws + 157815296u);
    float* shift = scale + 64;
    float* y     = (float*)(ws + 0);          // reuse h
    float* z2    = (float*)(ws + 73403904u);  // reuse x1u
    float* out   = (float*)d_out;

    gemm_up_kernel<<<dim3(2561, 7), 128, 0, stream>>>(x1, Wup, bup, h);
    build_x1u_kernel<<<40961, 256, 0, stream>>>(h, top, down, x1u);

    conv1_gemm_kernel<<<10241, 128, 16 * C1_STRIDE * 4, stream>>>(x1u, x2, ngh, W1, b1, z1);
    stats_partial_kernel<<<1024, 256, 0, stream>>>(z1, part);
    reduce_finalize_kernel<<<1, 256, 0, stream>>>(part, g1, be1, scale, shift);
    apply_bn_lrelu_kernel<<<10241, 256, 0, stream>>>(z1, scale, shift, y);

    conv2_gemm_kernel<<<10241, 128, 16 * C2_STRIDE * 4, stream>>>(y, ngh, W2, b2, z2);
    stats_partial_kernel<<<1024, 256, 0, stream>>>(z2, part);
    reduce_finalize_kernel<<<1, 256, 0, stream>>>(part, g2, be2, scale, shift);
    apply_bn_lrelu_kernel<<<10241, 256, 0, stream>>>(z2, scale, shift, out);
}